// GraphAttention_5935644803685
// MI455X (gfx1250) — compile-verified
//
#include <hip/hip_runtime.h>
#include <hip/hip_bf16.h>

// ---------------------------------------------------------------------------
// GAT layer for MI455X (gfx1250, wave32).
//   K1: H = X @ W via V_WMMA_F32_16X16X4_F32, 2 M-tiles per wave (B reuse),
//       branch-free store epilogue on the aligned fast path
//   K2: per-node attention scalars a_src/a_dst (collapses 256-wide edge dots)
//   K3: CSR row_ptr by binary search over sorted src
//   K4: one wave32 per node: fused score + softmax-normalized aggregation,
//       no atomics, deterministic, float4-coalesced gathers (L2-resident H)
// ---------------------------------------------------------------------------

typedef __attribute__((ext_vector_type(2))) float v2f;
typedef __attribute__((ext_vector_type(8))) float v8f;

#define D_FEAT 128
#define UNITS  128

__device__ __forceinline__ v8f wmma_f32_k4(v2f a, v2f b, v8f c) {
  return __builtin_amdgcn_wmma_f32_16x16x4_f32(
      /*neg_a=*/false, a, /*neg_b=*/false, b,
      /*c_mod=*/(short)0, c, /*reuse_a=*/false, /*reuse_b=*/false);
}

// ---- K1: H = X @ W. One wave owns a 32x16 C block (two 16x16 WMMA tiles),
//      sharing each B fragment across both M-tiles. Block = 8 waves = 8 N-tiles.
__global__ void gat_gemm_wmma(const float* __restrict__ X,
                              const float* __restrict__ W,
                              float* __restrict__ H, int n_nodes) {
  const int wave = threadIdx.x >> 5;        // 0..7 -> N tile
  const int lane = threadIdx.x & 31;
  const int half = lane >> 4;               // 0: lanes 0-15, 1: lanes 16-31
  const int l16  = lane & 15;
  const int Mbase = blockIdx.x * 32;
  const int Nbase = wave * 16;

  // A rows this lane feeds (lanes L and L+16 both carry row M = L within tile)
  const int rmax = n_nodes - 1;
  int r0 = Mbase + l16;       if (r0 > rmax) r0 = rmax;   // clamp (tail only)
  int r1 = Mbase + 16 + l16;  if (r1 > rmax) r1 = rmax;
  const float* __restrict__ A0   = X + (size_t)r0 * D_FEAT;
  const float* __restrict__ A1   = X + (size_t)r1 * D_FEAT;
  const float* __restrict__ Bcol = W + Nbase + l16;       // column N = Nbase+l16

  v8f c0 = {}, c1 = {};
#pragma unroll 4
  for (int k0 = 0; k0 < D_FEAT; k0 += 4) {
    const int ka = k0 + half * 2;  // lanes 0-15: K=k0,k0+1 ; lanes 16-31: K=k0+2,k0+3
    v2f a0, a1, b;
    a0.x = A0[ka];  a0.y = A0[ka + 1];
    a1.x = A1[ka];  a1.y = A1[ka + 1];
    b.x  = Bcol[(size_t)ka * UNITS];          // B[K][N], same half-lane K split
    b.y  = Bcol[(size_t)(ka + 1) * UNITS];
    c0 = wmma_f32_k4(a0, b, c0);              // B fragment reused by both tiles
    c1 = wmma_f32_k4(a1, b, c1);
  }

  // C layout: VGPR r holds row (r + half*8), col N = l16
  float* __restrict__ Hout = H + Nbase + l16;
  if (Mbase + 32 <= n_nodes) {                // uniform scalar test: fast path
#pragma unroll
    for (int r = 0; r < 8; ++r) {
      const int m = Mbase + r + half * 8;
      Hout[(size_t)m * UNITS]        = c0[r];
      Hout[(size_t)(m + 16) * UNITS] = c1[r];
    }
  } else {                                    // ragged tail tile only
#pragma unroll
    for (int r = 0; r < 8; ++r) {
      const int m = Mbase + r + half * 8;
      if (m < n_nodes)      Hout[(size_t)m * UNITS]        = c0[r];
      if (m + 16 < n_nodes) Hout[(size_t)(m + 16) * UNITS] = c1[r];
    }
  }
}

// ---- K2: a_src[i] = h[i]·ka[0:128], a_dst[i] = h[i]·ka[128:256] -----------
__global__ void gat_node_attention(const float* __restrict__ H,
                                   const float* __restrict__ KA,
                                   float* __restrict__ a_src,
                                   float* __restrict__ a_dst, int n_nodes) {
  const int wave = threadIdx.x >> 5;
  const int lane = threadIdx.x & 31;
  const int node = blockIdx.x * 8 + wave;
  if (node >= n_nodes) return;

  const float4 h4 = ((const float4*)(H + (size_t)node * UNITS))[lane];
  const float4 k0 = ((const float4*)KA)[lane];        // ka rows 0..127
  const float4 k1 = ((const float4*)KA)[32 + lane];   // ka rows 128..255
  float s0 = h4.x * k0.x + h4.y * k0.y + h4.z * k0.z + h4.w * k0.w;
  float s1 = h4.x * k1.x + h4.y * k1.y + h4.z * k1.z + h4.w * k1.w;
#pragma unroll
  for (int off = 16; off > 0; off >>= 1) {
    s0 += __shfl_xor(s0, off, 32);
    s1 += __shfl_xor(s1, off, 32);
  }
  if (lane == 0) { a_src[node] = s0; a_dst[node] = s1; }
}

// ---- K3: row_ptr[i] = lower_bound(src, i) over sorted src -----------------
__global__ void gat_rowptr(const long long* __restrict__ edges, int n_edges,
                           int n_nodes, int* __restrict__ row_ptr) {
  const int i = blockIdx.x * blockDim.x + threadIdx.x;
  if (i > n_nodes) return;
  int lo = 0, hi = n_edges;
  while (lo < hi) {
    const int mid = (lo + hi) >> 1;
    if (edges[2 * (size_t)mid] < (long long)i) lo = mid + 1; else hi = mid;
  }
  row_ptr[i] = lo;
}

// ---- K4: fused per-node score + aggregation (one wave32 per node) ---------
__global__ void gat_aggregate(const float* __restrict__ H,
                              const long long* __restrict__ edges,
                              const float* __restrict__ a_src,
                              const float* __restrict__ a_dst,
                              const int* __restrict__ row_ptr,
                              float* __restrict__ out, int n_nodes) {
  const int wave = threadIdx.x >> 5;
  const int lane = threadIdx.x & 31;
  const int node = blockIdx.x * 8 + wave;
  if (node >= n_nodes) return;

  const int es = row_ptr[node];
  const int ee = row_ptr[node + 1];
  const float asrc = a_src[node];

  float4 acc = make_float4(0.f, 0.f, 0.f, 0.f);
  float denom = 0.f;
  for (int j = es; j < ee; ++j) {
    const int d = (int)edges[2 * (size_t)j + 1];
    float logit = asrc + a_dst[d];                 // scalar broadcast load
    logit = (logit > 0.f) ? logit : 0.2f * logit;  // leaky_relu(0.2)
    logit = fminf(2.f, fmaxf(-2.f, logit));        // clip
    const float sc = __expf(logit);
    denom += sc;
    const float4 hv = ((const float4*)(H + (size_t)d * UNITS))[lane];
    acc.x += sc * hv.x; acc.y += sc * hv.y;
    acc.z += sc * hv.z; acc.w += sc * hv.w;
  }
  const float inv = (ee > es) ? (1.f / denom) : 0.f;  // empty segment -> zeros
  float4 o = make_float4(acc.x * inv, acc.y * inv, acc.z * inv, acc.w * inv);
  ((float4*)(out + (size_t)node * UNITS))[lane] = o;
}

// ---------------------------------------------------------------------------
extern "C" void kernel_launch(void* const* d_in, const int* in_sizes, int n_in,
                              void* d_out, int out_size, void* d_ws, size_t ws_size,
                              hipStream_t stream) {
  const float*     X  = (const float*)d_in[0];       // (N_NODES, 128)
  const long long* E  = (const long long*)d_in[1];   // (N_EDGES, 2) int64
  const float*     W  = (const float*)d_in[2];       // (128, 128)
  const float*     KA = (const float*)d_in[3];       // (256, 1)

  const int n_nodes = in_sizes[0] / D_FEAT;
  const int n_edges = in_sizes[1] / 2;

  // Workspace layout (bytes): H | a_src | a_dst | row_ptr
  char* ws = (char*)d_ws;
  float* H      = (float*)ws;                                  // n_nodes*128 f32
  float* a_src  = (float*)(ws + (size_t)n_nodes * UNITS * 4);
  float* a_dst  = a_src + n_nodes;
  int*   rowptr = (int*)(a_dst + n_nodes);                     // n_nodes+1 ints
  float* out    = (float*)d_out;

  const int mblocks = (n_nodes + 31) / 32;   // 32 rows (2 M-tiles) per block
  gat_gemm_wmma<<<mblocks, 256, 0, stream>>>(X, W, H, n_nodes);

  const int nblk8 = (n_nodes + 7) / 8;
  gat_node_attention<<<nblk8, 256, 0, stream>>>(H, KA, a_src, a_dst, n_nodes);

  gat_rowptr<<<(n_nodes + 256) / 256, 256, 0, stream>>>(E, n_edges, n_nodes, rowptr);

  gat_aggregate<<<nblk8, 256, 0, stream>>>(H, E, a_src, a_dst, rowptr, out, n_nodes);
}